// GraphModule_72095321030988
// MI455X (gfx1250) — compile-verified
//
#include <hip/hip_runtime.h>

typedef __attribute__((ext_vector_type(2))) float v2f;
typedef __attribute__((ext_vector_type(8))) float v8f;

#define NEG_SLOPE 0.2f

// float atomic max via signed/unsigned int ordering trick (works for mixed signs,
// memory initialized to -inf = 0xFF800000)
static __device__ __forceinline__ void atomic_max_f32(float* addr, float v) {
  if (v >= 0.0f) atomicMax((int*)addr, __float_as_int(v));
  else           atomicMin((unsigned int*)addr, __float_as_uint(v));
}

// ---------------------------------------------------------------- init kernels
__global__ void k_init_nodes(float* __restrict__ m, float* __restrict__ denom,
                             float* __restrict__ wsum, int N) {
  int i = blockIdx.x * blockDim.x + threadIdx.x;
  if (i < N) { m[i] = -__builtin_inff(); denom[i] = 0.0f; wsum[i] = 0.0f; }
}

__global__ void k_zero(float* __restrict__ p, int n) {
  int i = blockIdx.x * blockDim.x + threadIdx.x;
  if (i < n) p[i] = 0.0f;
}

// ------------------------------------------------------- fused GEMM + att dots
// h = x @ W  (x: [N,128] row-major, W: [128,64] row-major, h: [N,64])
// a_s = h . att_src, a_d = h . att_dst  (computed in epilogue via lane shuffles)
// One wave per 16-row tile; 4 f32 WMMA accumulator tiles cover the 64 columns.
//
// W is staged into LDS in k-pair interleaved layout:
//   Wl[(k/2)*128 + n*2 + (k&1)] = W[k][n]
// so a B fragment (W[kb][n], W[kb+1][n]) is ONE contiguous 8-byte ds_load_b64
// that lands directly in an even-aligned VGPR pair (no mov shuffling).
__global__ __launch_bounds__(128)
void k_gemm_att(const float* __restrict__ x, const float* __restrict__ Wg,
                const float* __restrict__ att_src, const float* __restrict__ att_dst,
                float* __restrict__ h, float* __restrict__ a_s, float* __restrict__ a_d,
                int N) {
  __shared__ float Wl[128 * 64];                       // 32 KB staged weights
  for (int i = threadIdx.x; i < 128 * 64; i += 128) {
    const int k = i >> 6, n = i & 63;
    Wl[(k >> 1) * 128 + n * 2 + (k & 1)] = Wg[i];
  }
  __syncthreads();

  const int lane = threadIdx.x & 31;
  const int wave = threadIdx.x >> 5;
  const int g    = lane >> 4;          // half-wave group (0/1)
  const int ln   = lane & 15;          // lane within group
  const int rowbase = (blockIdx.x * 4 + wave) * 16;
  if (rowbase >= N) return;            // uniform per wave: EXEC stays all-ones

  v8f c0 = {}, c1 = {}, c2 = {}, c3 = {};
  // A fragment source: lane ln reads row (rowbase+ln), K = 4*kk + 2*g + {0,1}
  const float* xr = x + (size_t)(rowbase + ln) * 128 + 2 * g;

#pragma unroll 8
  for (int kk = 0; kk < 32; ++kk) {
    v2f a;
    a.x = xr[kk * 4 + 0];
    a.y = xr[kk * 4 + 1];
    // B fragments: kb = 4*kk + 2*g  ->  pair row index kb/2 = 2*kk + g
    const v2f* wp = (const v2f*)&Wl[(kk * 2 + g) * 128];
    v2f b0 = wp[ 0 + ln];
    v2f b1 = wp[16 + ln];
    v2f b2 = wp[32 + ln];
    v2f b3 = wp[48 + ln];
    c0 = __builtin_amdgcn_wmma_f32_16x16x4_f32(false, a, false, b0, (short)0, c0, false, false);
    c1 = __builtin_amdgcn_wmma_f32_16x16x4_f32(false, a, false, b1, (short)0, c1, false, false);
    c2 = __builtin_amdgcn_wmma_f32_16x16x4_f32(false, a, false, b2, (short)0, c2, false, false);
    c3 = __builtin_amdgcn_wmma_f32_16x16x4_f32(false, a, false, b3, (short)0, c3, false, false);
  }

  // attention vectors per lane column
  const float as0 = att_src[ 0 + ln], as1 = att_src[16 + ln];
  const float as2 = att_src[32 + ln], as3 = att_src[48 + ln];
  const float ad0 = att_dst[ 0 + ln], ad1 = att_dst[16 + ln];
  const float ad2 = att_dst[32 + ln], ad3 = att_dst[48 + ln];

  // C layout: VGPR r -> row (rowbase + r + 8*g), column ntile*16 + ln
#pragma unroll
  for (int r = 0; r < 8; ++r) {
    const int row = rowbase + r + 8 * g;
    float* hp = h + (size_t)row * 64 + ln;
    hp[0]  = c0[r];
    hp[16] = c1[r];
    hp[32] = c2[r];
    hp[48] = c3[r];
    float ps = c0[r] * as0 + c1[r] * as1 + c2[r] * as2 + c3[r] * as3;
    float pd = c0[r] * ad0 + c1[r] * ad1 + c2[r] * ad2 + c3[r] * ad3;
#pragma unroll
    for (int off = 8; off; off >>= 1) {
      ps += __shfl_xor(ps, off, 16);
      pd += __shfl_xor(pd, off, 16);
    }
    if (ln == 0) { a_s[row] = ps; a_d[row] = pd; }
  }
}

// ------------------------------------------------------------------ edge passes
// Pass 1: e = leaky_relu(a_s[src] + a_d[dst]); store e; atomic-max into m[dst]
__global__ void k_edge_max(const long long* __restrict__ ei,
                           const float* __restrict__ a_s, const float* __restrict__ a_d,
                           float* __restrict__ m, float* __restrict__ e_out,
                           int E, int ET) {
  int t = blockIdx.x * blockDim.x + threadIdx.x;
  if (t >= ET) return;
  int s, d;
  if (t < E) { s = (int)ei[t]; d = (int)ei[E + t]; }
  else       { s = d = t - E; }                      // self-loop
  float e = a_s[s] + a_d[d];
  e = (e > 0.0f) ? e : NEG_SLOPE * e;
  e_out[t] = e;
  atomic_max_f32(&m[d], e);
}

// Pass 2: ex = exp(e - m[dst]); denom[dst] += ex; overwrite e_out with ex
__global__ void k_edge_exp(const long long* __restrict__ ei,
                           const float* __restrict__ m, float* __restrict__ e_out,
                           float* __restrict__ denom, int E, int ET) {
  int t = blockIdx.x * blockDim.x + threadIdx.x;
  if (t >= ET) return;
  int d = (t < E) ? (int)ei[E + t] : (t - E);
  float ex = __expf(e_out[t] - m[d]);
  e_out[t] = ex;
  atomicAdd(&denom[d], ex);
}

// Pass 3: alpha = ex / denom[dst]; wsum[src] += alpha
__global__ void k_edge_alpha(const long long* __restrict__ ei,
                             const float* __restrict__ e_out, const float* __restrict__ denom,
                             float* __restrict__ wsum, int E, int ET) {
  int t = blockIdx.x * blockDim.x + threadIdx.x;
  if (t >= ET) return;
  int s, d;
  if (t < E) { s = (int)ei[t]; d = (int)ei[E + t]; }
  else       { s = d = t - E; }
  float alpha = e_out[t] / denom[d];
  atomicAdd(&wsum[s], alpha);
}

// ------------------------------------------------- node reduction (streaming)
// acc[f] += sum_n wsum[n] * h[n][f]   (thread = (node lane, feature))
__global__ __launch_bounds__(256)
void k_node_reduce(const float* __restrict__ h, const float* __restrict__ wsum,
                   float* __restrict__ out_acc, int N) {
  const int f  = threadIdx.x & 63;
  const int nl = threadIdx.x >> 6;                   // 0..3
  float acc = 0.0f;
  for (int n = blockIdx.x * 4 + nl; n < N; n += gridDim.x * 4)
    acc += wsum[n] * h[(size_t)n * 64 + f];
  __shared__ float red[256];
  red[threadIdx.x] = acc;
  __syncthreads();
  if (threadIdx.x < 64) {
    float s = red[threadIdx.x] + red[threadIdx.x + 64] +
              red[threadIdx.x + 128] + red[threadIdx.x + 192];
    atomicAdd(&out_acc[threadIdx.x], s);
  }
}

__global__ void k_finalize(const float* __restrict__ out_acc, const float* __restrict__ bias,
                           float* __restrict__ out, int total, float invN) {
  int i = blockIdx.x * blockDim.x + threadIdx.x;
  if (i < total) out[i] = out_acc[i] * invN + bias[i & 63];
}

// ------------------------------------------------------------------- launcher
extern "C" void kernel_launch(void* const* d_in, const int* in_sizes, int n_in,
                              void* d_out, int out_size, void* d_ws, size_t ws_size,
                              hipStream_t stream) {
  const float*     node_input = (const float*)d_in[0];       // [B,N,128]
  const long long* edge_index = (const long long*)d_in[1];   // [2,E] int64
  const float*     W          = (const float*)d_in[2];       // [128,64]
  const float*     att_src    = (const float*)d_in[3];       // [64]
  const float*     att_dst    = (const float*)d_in[4];       // [64]
  const float*     bias       = (const float*)d_in[5];       // [64]
  float*           out        = (float*)d_out;               // [B,64]

  const int Fout = 64, Fin = 128;
  const int B  = out_size / Fout;                            // 4
  const int E  = in_sizes[1] / 2;                            // 800000
  const int N  = in_sizes[0] / (B * Fin);                    // 50000
  const int ET = E + N;                                      // edges + self-loops

  // workspace carve (processed batch-by-batch; ~17 MB total)
  char* ws = (char*)d_ws;
  float* h       = (float*)ws;  ws += (size_t)N * 64 * sizeof(float);
  float* a_s     = (float*)ws;  ws += (size_t)N * sizeof(float);
  float* a_d     = (float*)ws;  ws += (size_t)N * sizeof(float);
  float* m       = (float*)ws;  ws += (size_t)N * sizeof(float);
  float* denom   = (float*)ws;  ws += (size_t)N * sizeof(float);
  float* wsum    = (float*)ws;  ws += (size_t)N * sizeof(float);
  float* e_buf   = (float*)ws;  ws += (size_t)ET * sizeof(float);
  float* out_acc = (float*)ws;  ws += (size_t)B * Fout * sizeof(float);

  const int nThreads = 256;
  const int nodeBlocks = (N + nThreads - 1) / nThreads;
  const int edgeBlocks = (ET + nThreads - 1) / nThreads;
  const int gemmBlocks = (N + 63) / 64;                      // 4 waves x 16 rows

  k_zero<<<(B * Fout + 255) / 256, 256, 0, stream>>>(out_acc, B * Fout);

  for (int b = 0; b < B; ++b) {
    const float* x = node_input + (size_t)b * N * Fin;
    k_init_nodes<<<nodeBlocks, nThreads, 0, stream>>>(m, denom, wsum, N);
    k_gemm_att<<<gemmBlocks, 128, 0, stream>>>(x, W, att_src, att_dst, h, a_s, a_d, N);
    k_edge_max<<<edgeBlocks, nThreads, 0, stream>>>(edge_index, a_s, a_d, m, e_buf, E, ET);
    k_edge_exp<<<edgeBlocks, nThreads, 0, stream>>>(edge_index, m, e_buf, denom, E, ET);
    k_edge_alpha<<<edgeBlocks, nThreads, 0, stream>>>(edge_index, e_buf, denom, wsum, E, ET);
    k_node_reduce<<<256, 256, 0, stream>>>(h, wsum, out_acc + b * Fout, N);
  }

  k_finalize<<<1, 256, 0, stream>>>(out_acc, bias, out, B * Fout, 1.0f / (float)N);
}